// PScan_71021579207409
// MI455X (gfx1250) — compile-verified
//
#include <hip/hip_runtime.h>

// ---------------------------------------------------------------------------
// Chunked parallel scan of 16x16 f32 matrix recurrence Y_t = A_t Y_{t-1} + X_t
// on gfx1250 using V_WMMA_F32_16X16X4_F32 (4-chained => full 16x16x16 matmul).
//
// Phase 1: per chunk (T=64 steps): P_chunk = prod A, Yend = local scan end.
// Phase 2: per sequence: exclusive scan of 32 chunk carries (1 wave).
// Phase 3: per chunk: Y_t = A_t Y_{t-1} + X_t seeded with the chunk carry.
//
// C/D -> B-operand relayout done with v_permlane16_swap_b32 (4 VALU ops,
// no LDS traffic) instead of ds_bpermute.
// ---------------------------------------------------------------------------

typedef float v2f __attribute__((ext_vector_type(2)));
typedef float v8f __attribute__((ext_vector_type(8)));

constexpr int  B_   = 4;
constexpr int  L_   = 2048;
constexpr int  C_   = 16;
constexpr int  SEQ  = B_ * C_;        // 64 independent sequences
constexpr int  NCH  = 32;             // chunks per sequence
constexpr int  T_   = L_ / NCH;       // 64 timesteps per chunk
constexpr long TSTRIDE = (long)C_ * 256;  // float stride between t's at fixed (b,c)
constexpr long SUM_STRIDE  = 2 * 256;     // per-chunk ws: [P(256), Yend(256)]
constexpr long CARRY_OFF   = (long)SEQ * NCH * SUM_STRIDE;

struct Frag4 { v2f f[4]; };  // A-operand or B-operand fragments (4 K-blocks)

// Exchange lanes 16..31 of `a` with lanes 0..15 of `b` (single VALU op).
__device__ __forceinline__ void swapHalf(float& a, float& b) {
  typedef unsigned v2u __attribute__((ext_vector_type(2)));
  v2u r = __builtin_amdgcn_permlane16_swap(
      __builtin_bit_cast(unsigned, a), __builtin_bit_cast(unsigned, b),
      /*fi=*/false, /*bound_ctrl=*/false);
  a = __builtin_bit_cast(float, r[0]);
  b = __builtin_bit_cast(float, r[1]);
}

// Full 16x16x16 f32 matmul: acc = A * B + acc, via 4x V_WMMA_F32_16X16X4_F32.
__device__ __forceinline__ v8f mm16(const Frag4& a, const Frag4& b, v8f acc) {
#pragma unroll
  for (int kb = 0; kb < 4; ++kb) {
    acc = __builtin_amdgcn_wmma_f32_16x16x4_f32(
        /*neg_a=*/false, a.f[kb], /*neg_b=*/false, b.f[kb],
        /*c_mod=*/(short)0, acc, /*reuse_a=*/false, /*reuse_b=*/false);
  }
  return acc;
}

// Convert 16x16 matrix from C/D layout (reg v holds rows {v | v+8} across
// lane halves) to B-operand layout (frag kb = rows {4k,4k+1 | lo}/{4k+2,4k+3 | hi}).
// Exactly 4 v_permlane16_swap_b32: swap(c[v], c[v+2]) yields both fragments.
__device__ __forceinline__ Frag4 cdToB(v8f c) {
  Frag4 b;
  float t0 = c[0], t2 = c[2]; swapHalf(t0, t2); b.f[0].x = t0; b.f[2].x = t2;
  float t1 = c[1], t3 = c[3]; swapHalf(t1, t3); b.f[0].y = t1; b.f[2].y = t3;
  float t4 = c[4], t6 = c[6]; swapHalf(t4, t6); b.f[1].x = t4; b.f[3].x = t6;
  float t5 = c[5], t7 = c[7]; swapHalf(t5, t7); b.f[1].y = t5; b.f[3].y = t7;
  return b;
}

// Row-major 16x16 (contiguous 256 floats) -> A-operand fragments.
// A frag kb: lanes 0-15 hold M[row=lane][4kb+0,1]; lanes 16-31 hold M[row][4kb+2,3].
__device__ __forceinline__ Frag4 loadA_rm(const float* p, int lane) {
  const int row = lane & 15;
  const int ko  = (lane & 16) ? 2 : 0;
  Frag4 a;
#pragma unroll
  for (int kb = 0; kb < 4; ++kb)
    a.f[kb] = *(const v2f*)(p + row * 16 + kb * 4 + ko);  // 8B-aligned b64 loads
  return a;
}

// Row-major -> C/D layout fragments.
__device__ __forceinline__ v8f loadCD_rm(const float* p, int lane) {
  const int col = lane & 15;
  const int r0  = (lane & 16) ? 8 : 0;
  v8f c;
#pragma unroll
  for (int v = 0; v < 8; ++v) c[v] = p[(r0 + v) * 16 + col];
  return c;
}

// Row-major -> B-operand fragments.
__device__ __forceinline__ Frag4 loadB_rm(const float* p, int lane) {
  const int col = lane & 15;
  const int r   = (lane & 16) ? 2 : 0;
  Frag4 b;
#pragma unroll
  for (int kb = 0; kb < 4; ++kb) {
    b.f[kb].x = p[(4 * kb + r) * 16 + col];
    b.f[kb].y = p[(4 * kb + r + 1) * 16 + col];
  }
  return b;
}

// C/D layout -> row-major store.
__device__ __forceinline__ void storeCD_rm(float* p, const v8f& c, int lane) {
  const int col = lane & 15;
  const int r0  = (lane & 16) ? 8 : 0;
#pragma unroll
  for (int v = 0; v < 8; ++v) p[(r0 + v) * 16 + col] = c[v];
}

__device__ __forceinline__ Frag4 zeroFrag() {
  Frag4 z;
#pragma unroll
  for (int kb = 0; kb < 4; ++kb) { z.f[kb].x = 0.f; z.f[kb].y = 0.f; }
  return z;
}

// ---------------------------------------------------------------------------
// Phase 1: each wave owns one chunk; compute P_chunk and local Yend.
// ---------------------------------------------------------------------------
__global__ __launch_bounds__(256) void pscan_phase1(const float* __restrict__ A,
                                                    const float* __restrict__ X,
                                                    float* __restrict__ ws) {
  const int  lane  = threadIdx.x & 31;
  const bool hi    = (lane & 16) != 0;
  const int  w     = blockIdx.x * (blockDim.x >> 5) + (threadIdx.x >> 5);
  const int  seq   = w / NCH;
  const int  chunk = w % NCH;
  const int  b     = seq / C_;
  const int  c     = seq % C_;
  const long base  = (((long)b * L_ + (long)chunk * T_) * C_ + c) * 256;

  const float* aPtr = A + base;
  const float* xPtr = X + base;

  // p = identity in B-operand layout, y = 0
  Frag4 p_b;
#pragma unroll
  for (int kb = 0; kb < 4; ++kb) {
    const int r0 = 4 * kb + (hi ? 2 : 0);
    p_b.f[kb].x = ((lane & 15) == r0)     ? 1.f : 0.f;
    p_b.f[kb].y = ((lane & 15) == r0 + 1) ? 1.f : 0.f;
  }
  Frag4 y_b = zeroFrag();
  const v8f zero = {};
  v8f p_cd = zero, y_cd = zero;

#pragma unroll 2
  for (int t = 0; t < T_; ++t) {
    __builtin_prefetch(aPtr + TSTRIDE, 0, 3);   // global_prefetch_b8, next step
    __builtin_prefetch(xPtr + TSTRIDE, 0, 3);
    Frag4 a = loadA_rm(aPtr, lane);
    v8f   x = loadCD_rm(xPtr, lane);
    p_cd = mm16(a, p_b, zero);   // P <- A_t * P
    y_cd = mm16(a, y_b, x);      // y <- A_t * y + X_t
    p_b  = cdToB(p_cd);
    y_b  = cdToB(y_cd);
    aPtr += TSTRIDE;
    xPtr += TSTRIDE;
  }

  float* slot = ws + (long)(seq * NCH + chunk) * SUM_STRIDE;
  storeCD_rm(slot,        p_cd, lane);
  storeCD_rm(slot + 256,  y_cd, lane);
}

// ---------------------------------------------------------------------------
// Phase 2: one wave per sequence; exclusive scan of the 32 chunk summaries:
//   carry_0 = 0 ; carry_{k+1} = Yend_k + P_k * carry_k
// ---------------------------------------------------------------------------
__global__ __launch_bounds__(32) void pscan_phase2(float* __restrict__ ws) {
  const int  lane = threadIdx.x & 31;
  const int  seq  = blockIdx.x;

  const float* sumBase   = ws + (long)seq * NCH * SUM_STRIDE;
  float*       carryBase = ws + CARRY_OFF + (long)seq * NCH * 256;

  v8f   carry_cd = {};
  Frag4 carry_b  = zeroFrag();

  for (int k = 0; k < NCH; ++k) {
    storeCD_rm(carryBase + (long)k * 256, carry_cd, lane);
    if (k + 1 < NCH) {
      Frag4 aP   = loadA_rm(sumBase + (long)k * SUM_STRIDE, lane);
      v8f   yEnd = loadCD_rm(sumBase + (long)k * SUM_STRIDE + 256, lane);
      carry_cd = mm16(aP, carry_b, yEnd);   // carry <- P_k * carry + Yend_k
      carry_b  = cdToB(carry_cd);
    }
  }
}

// ---------------------------------------------------------------------------
// Phase 3: each wave replays its chunk seeded with the carry, writing Y.
// ---------------------------------------------------------------------------
__global__ __launch_bounds__(256) void pscan_phase3(const float* __restrict__ A,
                                                    const float* __restrict__ X,
                                                    const float* __restrict__ ws,
                                                    float* __restrict__ Y) {
  const int  lane  = threadIdx.x & 31;
  const int  w     = blockIdx.x * (blockDim.x >> 5) + (threadIdx.x >> 5);
  const int  seq   = w / NCH;
  const int  chunk = w % NCH;
  const int  b     = seq / C_;
  const int  c     = seq % C_;
  const long base  = (((long)b * L_ + (long)chunk * T_) * C_ + c) * 256;

  const float* aPtr = A + base;
  const float* xPtr = X + base;
  float*       yPtr = Y + base;

  const float* carry = ws + CARRY_OFF + (long)(seq * NCH + chunk) * 256;
  Frag4 y_b = loadB_rm(carry, lane);   // Y_{chunk_start - 1}

#pragma unroll 2
  for (int t = 0; t < T_; ++t) {
    __builtin_prefetch(aPtr + TSTRIDE, 0, 3);
    __builtin_prefetch(xPtr + TSTRIDE, 0, 3);
    Frag4 a = loadA_rm(aPtr, lane);
    v8f   x = loadCD_rm(xPtr, lane);
    v8f y_cd = mm16(a, y_b, x);        // Y_t = A_t * Y_{t-1} + X_t
    storeCD_rm(yPtr, y_cd, lane);
    y_b = cdToB(y_cd);
    aPtr += TSTRIDE;
    xPtr += TSTRIDE;
    yPtr += TSTRIDE;
  }
}

// ---------------------------------------------------------------------------
extern "C" void kernel_launch(void* const* d_in, const int* in_sizes, int n_in,
                              void* d_out, int out_size, void* d_ws, size_t ws_size,
                              hipStream_t stream) {
  const float* A  = (const float*)d_in[0];
  const float* X  = (const float*)d_in[1];
  float*       Y  = (float*)d_out;
  float*       ws = (float*)d_ws;   // needs SEQ*NCH*3*256 floats = 6 MiB

  const int wavesTotal = SEQ * NCH;           // 2048 chunk-waves
  const int blk1       = wavesTotal / 8;      // 8 waves per 256-thread block

  pscan_phase1<<<blk1, 256, 0, stream>>>(A, X, ws);
  pscan_phase2<<<SEQ, 32, 0, stream>>>(ws);
  pscan_phase3<<<blk1, 256, 0, stream>>>(A, X, ws, Y);
}